// ReEigLayer_34754875359644
// MI455X (gfx1250) — compile-verified
//
#include <hip/hip_runtime.h>

#define REEIG_EPS   1e-4f
#define REEIG_CLAMP 1e6f
#define NSWEEPS     8
#define WPB         4      // waves (independent matrices) per block

typedef float v2f __attribute__((ext_vector_type(2)));
typedef float v8f __attribute__((ext_vector_type(8)));

// One matrix per wave32; LDS used only for WMMA staging. Cross-lane LDS deps
// need only compiler-order enforcement (intra-wave DS ops complete in order).
static __device__ __forceinline__ void wave_sync() {
    __builtin_amdgcn_fence(__ATOMIC_ACQ_REL, "wavefront");
    __builtin_amdgcn_wave_barrier();
    __builtin_amdgcn_fence(__ATOMIC_ACQ_REL, "wavefront");
}

__global__ __launch_bounds__(WPB * 32)
void reeig_hestenes_wmma_kernel(const float* __restrict__ X,
                                float* __restrict__ out,
                                int batches) {
    // staging only: stride 33 floats -> conflict-free rows and columns
    __shared__ float BufA[WPB][32 * 33];   // G1 = g * (wr/lambda)
    __shared__ float BufB[WPB][32 * 33];   // G2 = g * (1/lambda) = V

    const int j  = threadIdx.x & 31;       // lane j owns column j of G
    const int w  = threadIdx.x >> 5;
    const int bm = blockIdx.x * WPB + w;
    if (bm >= batches) return;             // whole wave exits together

    const float* Xb = X   + (size_t)bm * 1024;
    float*       Ob = out + (size_t)bm * 1024;

    // ---- G = X + eps*I, column j in registers (coalesced loads) ----
    float g[32];
#pragma unroll
    for (int i = 0; i < 32; ++i) {
        float x = Xb[i * 32 + j];
        g[i] = (i == j) ? (x + REEIG_EPS) : x;
    }

    // ---- one-sided (Hestenes) Jacobi: orthogonalize column pairs ----
    for (int sweep = 0; sweep < NSWEEPS; ++sweep) {
        for (int r = 0; r < 31; ++r) {
            // round-robin tournament pairing: player 31 fixed, others rotate
            int pj;
            if (j == 31) {
                pj = r;
            } else {
                int pos = j - r; if (pos < 0) pos += 31;
                pj = (pos == 0) ? 31 : (31 - pos + r) % 31;
            }
            const bool isP = j < pj;

            // fetch partner column (shuffles) + Gram entries, 4-way accumulators
            float gp[32];
            float a0 = 0.f, a1 = 0.f, a2 = 0.f, a3 = 0.f;   // <g_j, g_pj>
            float o0 = 0.f, o1 = 0.f, o2 = 0.f, o3 = 0.f;   // <g_j, g_j>
#pragma unroll
            for (int i = 0; i < 32; i += 4) {
                gp[i + 0] = __shfl(g[i + 0], pj, 32);
                gp[i + 1] = __shfl(g[i + 1], pj, 32);
                gp[i + 2] = __shfl(g[i + 2], pj, 32);
                gp[i + 3] = __shfl(g[i + 3], pj, 32);
                a0 = fmaf(g[i + 0], gp[i + 0], a0);
                a1 = fmaf(g[i + 1], gp[i + 1], a1);
                a2 = fmaf(g[i + 2], gp[i + 2], a2);
                a3 = fmaf(g[i + 3], gp[i + 3], a3);
                o0 = fmaf(g[i + 0], g[i + 0], o0);
                o1 = fmaf(g[i + 1], g[i + 1], o1);
                o2 = fmaf(g[i + 2], g[i + 2], o2);
                o3 = fmaf(g[i + 3], g[i + 3], o3);
            }
            const float dpq  = (a0 + a1) + (a2 + a3);   // bit-identical on both pair lanes
            const float down = (o0 + o1) + (o2 + o3);
            const float dprt = __shfl(down, pj, 32);
            const float dpp  = isP ? down : dprt;
            const float dqq  = isP ? dprt : down;

            // Jacobi rotation zeroing the Gram off-diagonal
            float c, s;
            if (fabsf(dpq) > 1e-30f) {
                float tau = (dqq - dpp) / (2.0f * dpq);
                float t   = copysignf(1.0f, tau) / (fabsf(tau) + sqrtf(1.0f + tau * tau));
                c = rsqrtf(1.0f + t * t);
                s = t * c;
            } else {
                c = 1.0f; s = 0.0f;
            }
            const float ssn = isP ? -s : s;   // sign-folded -> branchless update

#pragma unroll
            for (int i = 0; i < 32; ++i)
                g[i] = fmaf(c, g[i], ssn * gp[i]);
        }
    }

    // ---- lambda_j = ||g_j||, v_j = g_j/lambda_j; rectify; stage for WMMA ----
    float n0 = 0.f, n1 = 0.f, n2 = 0.f, n3 = 0.f;
#pragma unroll
    for (int i = 0; i < 32; i += 4) {
        n0 = fmaf(g[i + 0], g[i + 0], n0);
        n1 = fmaf(g[i + 1], g[i + 1], n1);
        n2 = fmaf(g[i + 2], g[i + 2], n2);
        n3 = fmaf(g[i + 3], g[i + 3], n3);
    }
    const float d   = (n0 + n1) + (n2 + n3);
    const float ir  = rsqrtf(d);                       // 1/lambda
    const float lam = d * ir;                          // lambda
    const float wr  = fminf(fmaxf(lam, REEIG_EPS), REEIG_CLAMP);
    const float s1  = wr * ir;                         // G1 scale: wr/lambda

    float* Ga = &BufA[w][0];
    float* Gv = &BufB[w][0];
#pragma unroll
    for (int i = 0; i < 32; ++i) {
        Ga[i * 33 + j] = g[i] * s1;                    // V * diag(wr), row-major
        Gv[i * 33 + j] = g[i] * ir;                    // V, row-major
    }
    wave_sync();

    // ---- out = G1 * G2^T via V_WMMA_F32_16X16X4_F32 (2x2 tiles, K=32 step 4) ----
    // A-frag 16x4 f32: lanes 0-15 (M=lane, K=k..k+1); lanes 16-31 (M=lane-16, K=k+2..k+3)
    // B-frag 4x16 f32: B[kk][n] = G2[n][k+kk]; same lane-half K split, N = lane%16
    const int half = j >> 4;
    const int lj   = j & 15;
#pragma unroll
    for (int ti = 0; ti < 2; ++ti) {
#pragma unroll
        for (int tj = 0; tj < 2; ++tj) {
            v8f acc = {0.0f, 0.0f, 0.0f, 0.0f, 0.0f, 0.0f, 0.0f, 0.0f};
#pragma unroll
            for (int k = 0; k < 32; k += 4) {
                v2f a, bf;
                const int ak = k + half * 2;
                a.x  = Ga[(ti * 16 + lj) * 33 + ak];
                a.y  = Ga[(ti * 16 + lj) * 33 + ak + 1];
                bf.x = Gv[(tj * 16 + lj) * 33 + ak];
                bf.y = Gv[(tj * 16 + lj) * 33 + ak + 1];
                acc = __builtin_amdgcn_wmma_f32_16x16x4_f32(
                    false, a, false, bf, (short)0, acc, false, false);
            }
            // C/D layout: VGPR d -> (M = 16*ti + 8*half + d, N = 16*tj + lane%16)
#pragma unroll
            for (int dd = 0; dd < 8; ++dd) {
                Ob[(ti * 16 + half * 8 + dd) * 32 + tj * 16 + lj] = acc[dd];
            }
        }
    }
}

extern "C" void kernel_launch(void* const* d_in, const int* in_sizes, int n_in,
                              void* d_out, int out_size, void* d_ws, size_t ws_size,
                              hipStream_t stream) {
    const float* X   = (const float*)d_in[0];
    float*       out = (float*)d_out;
    const int batches = in_sizes[0] / 1024;       // B x 32 x 32
    const int blocks  = (batches + WPB - 1) / WPB;
    hipLaunchKernelGGL(reeig_hestenes_wmma_kernel,
                       dim3(blocks), dim3(WPB * 32), 0, stream,
                       X, out, batches);
}